// UncertaintyMixedCoding_41077067219222
// MI455X (gfx1250) — compile-verified
//
#include <hip/hip_runtime.h>
#include <hip/hip_bf16.h>
#include <math.h>

// Problem constants (from reference setup_inputs)
#define BI 64
#define RR 36
#define DD 512
#define BC 64
#define WW 50
#define RP 48            // rows padded to 3 x 16 WMMA M-tiles
#define WP 64            // words padded to 4 x 16 WMMA N-tiles
#define ISTRIDE 516      // 516 % 64 == 4 -> bank = 4*row + k, conflict-free fragments
#define CSTRIDE 516
#define SSTRIDE 68       // 68 % 64 == 4
#define MASK_VAL (-1.0f)
#define EPSV 1e-12f
#define INV_T 10.0f      // 1 / TEMPERATURE

typedef __attribute__((ext_vector_type(2))) float v2f;
typedef __attribute__((ext_vector_type(8))) float v8f;

// ---------------------------------------------------------------------------
// Kernel 1: per image-region precompute
//   unc[i,r]  = sigmoid(relu(imgs[i,r,:] @ W1 + b1) @ W2 + b2)
//   invn[i,r] = 1 / max(||imgs[i,r,:]||, 1e-12)
// ---------------------------------------------------------------------------
__global__ __launch_bounds__(256)
void umc_prep_kernel(const float* __restrict__ imgs,
                     const float* __restrict__ W1, const float* __restrict__ b1,
                     const float* __restrict__ W2, const float* __restrict__ b2,
                     float* __restrict__ unc, float* __restrict__ invn)
{
    __shared__ float redA[256];
    __shared__ float redB[256];
    const int row = blockIdx.x;                 // i*RR + r
    const int t = threadIdx.x;
    const float* x = imgs + (size_t)row * DD;

    float acc = 0.0f;
    for (int d = 0; d < DD; ++d)
        acc = fmaf(x[d], W1[d * 256 + t], acc);
    float h = fmaxf(acc + b1[t], 0.0f);

    float xa = x[t];
    float xb = x[t + 256];
    redA[t] = h * W2[t];
    redB[t] = xa * xa + xb * xb;
    __syncthreads();
    for (int s = 128; s > 0; s >>= 1) {
        if (t < s) { redA[t] += redA[t + s]; redB[t] += redB[t + s]; }
        __syncthreads();
    }
    if (t == 0) {
        float z = redA[0] + b2[0];
        unc[row] = 1.0f / (1.0f + __expf(-z));
        invn[row] = 1.0f / fmaxf(sqrtf(redB[0]), EPSV);
    }
}

// Issue one 16-byte async global->LDS copy (CDNA5 GLOBAL_LOAD_ASYNC_TO_LDS_B128,
// tracked on ASYNCcnt). ldsPtr: generic pointer into shared memory; per ISA the
// low 32 bits of the flat address are the LDS address.
__device__ __forceinline__ void async_copy_b128(void* ldsPtr, const void* gPtr) {
    unsigned l = (unsigned)(size_t)ldsPtr;
    asm volatile("global_load_async_to_lds_b128 %0, %1, off"
                 :: "v"(l), "v"(gPtr) : "memory");
}

// ---------------------------------------------------------------------------
// Kernel 2: fused attention block per (i, j). 128 threads = 4 waves.
//   Phase A: S = imgs_i @ caps_j^T  (48x64, K=512) via v_wmma_f32_16x16x4_f32
//   Attn   : masked argmax/softmax + uncertainty mixing, in-place in LDS
//   Phase B: WC = P @ caps_j (K=64, N=512), folded into per-row sums
//   Out    : cosine(imgs_i[r], WC[r]) with img-length masking
// ---------------------------------------------------------------------------
__global__ __launch_bounds__(128)
void umc_fused_kernel(const float* __restrict__ imgs,
                      const float* __restrict__ caps,
                      const int* __restrict__ img_lens,
                      const int* __restrict__ cap_lens,
                      const float* __restrict__ unc,
                      const float* __restrict__ invn,
                      float* __restrict__ out)
{
    extern __shared__ float smem[];
    float* sImg = smem;                        // RP * ISTRIDE
    float* sCap = sImg + RP * ISTRIDE;         // WP * CSTRIDE
    float* sS   = sCap + WP * CSTRIDE;         // RP * SSTRIDE (sims -> attn)
    float* sSq  = sS + RP * SSTRIDE;           // RP : sum wc^2 per row
    float* sCr  = sSq + RP;                    // RP : sum wc*img per row

    const int i = blockIdx.x >> 6;
    const int j = blockIdx.x & 63;
    const int tid  = threadIdx.x;
    const int wave = tid >> 5;
    const int lane = tid & 31;
    const int half = lane >> 4;                // 0: K{0,1} lanes, 1: K{2,3} lanes
    const int l16  = lane & 15;

    // --- Zero the padding rows (they feed WMMA fragments; must not be NaN).
    const float4 z4 = make_float4(0.f, 0.f, 0.f, 0.f);
    for (int idx = tid; idx < (RP - RR) * (DD / 4); idx += 128) {
        int r = RR + (idx >> 7), d4 = idx & 127;
        *(float4*)&sImg[r * ISTRIDE + 4 * d4] = z4;
    }
    for (int idx = tid; idx < (WP - WW) * (DD / 4); idx += 128) {
        int w = WW + (idx >> 7), d4 = idx & 127;
        *(float4*)&sCap[w * CSTRIDE + 4 * d4] = z4;
    }
    if (tid < RP) { sSq[tid] = 0.0f; sCr[tid] = 0.0f; }

    // --- Async-stage imgs[i] (36x512) and caps[j] (50x512) straight into LDS.
    const float* gI = imgs + (size_t)i * RR * DD;
    for (int idx = tid; idx < RR * (DD / 4); idx += 128) {
        int r = idx >> 7, d4 = idx & 127;
        async_copy_b128(&sImg[r * ISTRIDE + 4 * d4], gI + r * DD + 4 * d4);
    }
    const float* gC = caps + (size_t)j * WW * DD;
    for (int idx = tid; idx < WW * (DD / 4); idx += 128) {
        int w = idx >> 7, d4 = idx & 127;
        async_copy_b128(&sCap[w * CSTRIDE + 4 * d4], gC + w * DD + 4 * d4);
    }
    asm volatile("s_wait_asynccnt 0" ::: "memory");
    __syncthreads();

    // --- Phase A: sims. Wave owns N-tile `wave`; loop 3 M-tiles; K=512 (128 steps).
    {
        const int n0 = wave * 16;
        const float* bRow = sCap + (n0 + l16) * CSTRIDE + 2 * half;  // B[k][n]=caps[n][k]
        for (int m = 0; m < 3; ++m) {
            const int r0 = m * 16;
            const float* aRow = sImg + (r0 + l16) * ISTRIDE + 2 * half;
            v8f acc = {};
            #pragma unroll 8
            for (int k = 0; k < DD; k += 4) {
                v2f a = *(const v2f*)(aRow + k);
                v2f b = *(const v2f*)(bRow + k);
                acc = __builtin_amdgcn_wmma_f32_16x16x4_f32(
                    false, a, false, b, (short)0, acc, false, false);
            }
            #pragma unroll
            for (int v = 0; v < 8; ++v)
                sS[(r0 + v + 8 * half) * SSTRIDE + n0 + l16] = acc[v];
        }
    }
    __syncthreads();

    // --- Attention: one thread per padded row, in-place S -> mixed attn P.
    const int iLen = img_lens[i];
    const int cLen = cap_lens[j];
    if (tid < RP) {
        const int r = tid;
        float* row = sS + r * SSTRIDE;
        const bool rValid = (r < RR) && (r < iLen);
        if (!rValid) {
            for (int w = 0; w < WP; ++w) row[w] = 0.0f;   // padded/masked rows: no attn
        } else {
            float smax = -INFINITY;
            float hbest = -INFINITY; int harg = 0;
            for (int w = 0; w < cLen; ++w) {
                float s = row[w];
                smax = fmaxf(smax, s);
                if (s > hbest) { hbest = s; harg = w; }
            }
            // jnp.argmax over where(mask, sims, -1.0): if every valid sim is below
            // the -1.0 mask value and masked slots exist, argmax hits a masked slot
            // and one_hot * mask kills the hard term entirely.
            const bool hardZero = (hbest < MASK_VAL) && (cLen < WW);
            float Z = 0.0f;
            for (int w = 0; w < cLen; ++w)
                Z += __expf((row[w] - smax) * INV_T);
            const float invZ = 1.0f / Z;
            const float alpha = unc[i * RR + r];
            for (int w = 0; w < WP; ++w) {
                float p = 0.0f;
                if (w < cLen) {
                    float soft = __expf((row[w] - smax) * INV_T) * invZ;
                    float hard = (!hardZero && w == harg) ? 1.0f : 0.0f;
                    p = alpha * soft + (1.0f - alpha) * hard;
                }
                row[w] = p;
            }
        }
    }
    __syncthreads();

    // --- Phase B: WC = P @ caps (K=64, N=512). Wave owns d-range [wave*128, +128).
    // Each 16x16 WC tile folds straight into per-row sq/cross accumulators.
    for (int m = 0; m < 3; ++m) {
        const int r0 = m * 16;
        const float* aRow = sS + (r0 + l16) * SSTRIDE + 2 * half;
        float sqv[8]; float crv[8];
        #pragma unroll
        for (int v = 0; v < 8; ++v) { sqv[v] = 0.0f; crv[v] = 0.0f; }

        for (int nt = 0; nt < 8; ++nt) {
            const int d0 = wave * 128 + nt * 16;
            v8f acc = {};
            #pragma unroll
            for (int k = 0; k < WP; k += 4) {
                v2f a = *(const v2f*)(aRow + k);
                v2f b;
                b.x = sCap[(k + 2 * half)     * CSTRIDE + d0 + l16];
                b.y = sCap[(k + 2 * half + 1) * CSTRIDE + d0 + l16];
                acc = __builtin_amdgcn_wmma_f32_16x16x4_f32(
                    false, a, false, b, (short)0, acc, false, false);
            }
            #pragma unroll
            for (int v = 0; v < 8; ++v) {
                float wc = acc[v];
                int rr = r0 + v + 8 * half;
                int dd = d0 + l16;
                sqv[v] = fmaf(wc, wc, sqv[v]);
                crv[v] = fmaf(wc, sImg[rr * ISTRIDE + dd], crv[v]);
            }
        }
        // Reduce across the 16 lanes (distinct d) sharing each row, then LDS-atomic.
        #pragma unroll
        for (int v = 0; v < 8; ++v) {
            float s = sqv[v], c = crv[v];
            #pragma unroll
            for (int off = 1; off < 16; off <<= 1) {
                s += __shfl_xor(s, off, 32);
                c += __shfl_xor(c, off, 32);
            }
            if (l16 == 0) {
                int rr = r0 + v + 8 * half;
                atomicAdd(&sSq[rr], s);
                atomicAdd(&sCr[rr], c);
            }
        }
    }
    __syncthreads();

    // --- Emit out[i][j][r]
    if (tid < RR) {
        const int r = tid;
        float res;
        if (r < iLen) {
            float invNw = 1.0f / fmaxf(sqrtf(sSq[r]), EPSV);
            res = sCr[r] * invn[i * RR + r] * invNw;
        } else {
            res = MASK_VAL;
        }
        out[((size_t)i * BC + j) * RR + r] = res;
    }
}

// ---------------------------------------------------------------------------
extern "C" void kernel_launch(void* const* d_in, const int* in_sizes, int n_in,
                              void* d_out, int out_size, void* d_ws, size_t ws_size,
                              hipStream_t stream) {
    const float* imgs     = (const float*)d_in[0];
    const float* caps     = (const float*)d_in[1];
    const int*   img_lens = (const int*)d_in[2];
    const int*   cap_lens = (const int*)d_in[3];
    const float* W1       = (const float*)d_in[4];
    const float* b1       = (const float*)d_in[5];
    const float* W2       = (const float*)d_in[6];
    const float* b2       = (const float*)d_in[7];
    float* out  = (float*)d_out;
    float* unc  = (float*)d_ws;            // BI*RR floats
    float* invn = unc + BI * RR;           // BI*RR floats

    umc_prep_kernel<<<BI * RR, 256, 0, stream>>>(imgs, W1, b1, W2, b2, unc, invn);

    const size_t shmem =
        (size_t)(RP * ISTRIDE + WP * CSTRIDE + RP * SSTRIDE + 2 * RP) * sizeof(float);
    (void)hipFuncSetAttribute((const void*)umc_fused_kernel,
                              hipFuncAttributeMaxDynamicSharedMemorySize,
                              (int)shmem);
    umc_fused_kernel<<<BI * BC, 128, shmem, stream>>>(
        imgs, caps, img_lens, cap_lens, unc, invn, out);
}